// SSIMLoss_53644141527299
// MI455X (gfx1250) — compile-verified
//
#include <hip/hip_runtime.h>

typedef float v2f __attribute__((ext_vector_type(2)));
typedef float v8f __attribute__((ext_vector_type(8)));

#define IMG   512
#define NBAT  32
#define KS    11
#define PAD   5
#define TILE  32
#define HALO  (TILE + KS - 1)    // 42 valid halo rows/cols
#define AROWS 48                 // padded halo rows stored (42..47 zero)
#define ASTR  44                 // padded halo col stride  (42..43 zero)
#define HROWS 48                 // H rows (42..47 are zero)
#define HSTR  36                 // padded LDS column stride for H (bank spread)
#define TPR   (IMG / TILE)       // 16 tiles per image row
#define TPI   (TPR * TPR)        // 256 tiles per image

__global__ void ssim_zero_kernel(float* acc) { acc[0] = 0.f; }

__global__ __launch_bounds__(128) void ssim_main_kernel(
    const float* __restrict__ xhat,
    const float* __restrict__ x,
    const float* __restrict__ k2d,
    float* __restrict__ acc)
{
  __shared__ float sA[AROWS * ASTR];         // a = (x + 1)/2, zero outside halo
  __shared__ float sB[AROWS * ASTR];         // b = (xhat + 1)/2
  __shared__ float sH[5][HROWS * HSTR];      // horizontal-pass results, 5 signals
  __shared__ float sg[16];                   // 1-D gaussian (11 taps)

  const int tid  = threadIdx.x;              // 0..127
  const int lane = tid & 31;
  const int wave = tid >> 5;                 // 0..3
  const int lhi  = lane >> 4;                // 0/1 (half-wave)
  const int llo  = lane & 15;

  const int bid  = blockIdx.x;
  const int img  = bid / TPI;
  const int t    = bid - img * TPI;
  const int ty   = t / TPR;
  const int tx   = t - ty * TPR;
  const int row0 = ty * TILE - PAD;
  const int col0 = tx * TILE - PAD;

  const float* px = x    + (size_t)img * IMG * IMG;
  const float* ph = xhat + (size_t)img * IMG * IMG;

  // Recover 1-D gaussian g from k2d = outer(g,g): g[j] = k2d[5][j] / sum(k2d[5][:])
  if (tid == 0) {
    float s = 0.f;
    for (int j = 0; j < KS; ++j) s += k2d[5 * KS + j];
    float inv = 1.0f / s;
    for (int j = 0; j < KS; ++j) sg[j] = k2d[5 * KS + j] * inv;
  }

  // Cooperative load: whole padded region; zeros outside the valid halo/image.
  for (int i = tid; i < AROWS * ASTR; i += 128) {
    int r  = i / ASTR;
    int c  = i - r * ASTR;
    int gr = row0 + r, gc = col0 + c;
    bool in = (r < HALO) && (c < HALO) &&
              (gr >= 0) && (gr < IMG) && (gc >= 0) && (gc < IMG);
    size_t gi = (size_t)gr * IMG + gc;
    sA[i] = in ? (px[gi] + 1.f) * 0.5f : 0.f;
    sB[i] = in ? (ph[gi] + 1.f) * 0.5f : 0.f;
  }
  __syncthreads();

  // ---------- Stage B: horizontal conv for all 5 signals at once ----------
  // D(16x16) = A(16x4 data) x B(4x16 Toeplitz), 7 K-chunks (K=28 >= 26 needed).
  v2f Bt[7];
#pragma unroll
  for (int c = 0; c < 7; ++c) {
#pragma unroll
    for (int v = 0; v < 2; ++v) {
      int k = 4 * c + 2 * lhi + v;       // K index (0..27)
      int d = k - llo;                   // tap index g[k - n]
      Bt[c][v] = (d >= 0 && d < KS) ? sg[d] : 0.f;
    }
  }

  // 6 (mC,nC) chunks over 4 waves: chunk = wave, and chunk = wave+4 for waves 0,1.
  auto hchunk = [&](int chunk) {
    const int mC = (chunk >= 3) ? chunk - 3 : chunk;   // 0..2 (row chunk)
    const int nC = (chunk >= 3) ? 1 : 0;               // 0..1 (col chunk)
    const int m0 = mC * 16;
    const int n0 = nC * 16;
    const int aBase = (m0 + llo) * ASTR + n0;          // always in-bounds (padded)

    v8f ca = {}, cb = {}, caa = {}, cbb = {}, cab = {};
#pragma unroll
    for (int c = 0; c < 7; ++c) {
      const int k0 = 4 * c + 2 * lhi;
      float a0 = sA[aBase + k0],     b0 = sB[aBase + k0];
      float a1 = sA[aBase + k0 + 1], b1 = sB[aBase + k0 + 1];
      v2f Aa  = {a0, b0 * 0.f + a0};   // (placeholder removed below)
      Aa[0] = a0; Aa[1] = a1;
      v2f Ab;  Ab[0] = b0;       Ab[1] = b1;
      v2f Aaa; Aaa[0] = a0 * a0; Aaa[1] = a1 * a1;
      v2f Abb; Abb[0] = b0 * b0; Abb[1] = b1 * b1;
      v2f Aab; Aab[0] = a0 * b0; Aab[1] = a1 * b1;
      ca  = __builtin_amdgcn_wmma_f32_16x16x4_f32(false, Aa,  false, Bt[c], (short)0, ca,  false, false);
      cb  = __builtin_amdgcn_wmma_f32_16x16x4_f32(false, Ab,  false, Bt[c], (short)0, cb,  false, false);
      caa = __builtin_amdgcn_wmma_f32_16x16x4_f32(false, Aaa, false, Bt[c], (short)0, caa, false, false);
      cbb = __builtin_amdgcn_wmma_f32_16x16x4_f32(false, Abb, false, Bt[c], (short)0, cbb, false, false);
      cab = __builtin_amdgcn_wmma_f32_16x16x4_f32(false, Aab, false, Bt[c], (short)0, cab, false, false);
    }
    // C layout: element v of lane l -> row m0 + v + 8*(l>>4), col n0 + (l&15)
    const int hBase = (m0 + 8 * lhi) * HSTR + n0 + llo;
#pragma unroll
    for (int v = 0; v < 8; ++v) {
      sH[0][hBase + v * HSTR] = ca[v];
      sH[1][hBase + v * HSTR] = cb[v];
      sH[2][hBase + v * HSTR] = caa[v];
      sH[3][hBase + v * HSTR] = cbb[v];
      sH[4][hBase + v * HSTR] = cab[v];
    }
  };
  hchunk(wave);
  if (wave < 2) hchunk(wave + 4);
  __syncthreads();

  // ---------- Stage C: vertical conv, Out[i][j] = sum_t g[t]*H[r0+i+t][j] ----------
  // D(16x16) = A(16x4 Toeplitz) x B(4x16 H-rows), 7 K-chunks, per-wave subtile.
  const int r0 = (wave >> 1) * 16;       // subtile row offset (H-row space)
  const int j0 = (wave & 1) * 16;        // subtile col offset

  v2f At[7];
#pragma unroll
  for (int c = 0; c < 7; ++c) {
#pragma unroll
    for (int v = 0; v < 2; ++v) {
      int k = 4 * c + 2 * lhi + v;       // K index (0..27)
      int d = k - llo;                   // tap index g[k - i]
      At[c][v] = (d >= 0 && d < KS) ? sg[d] : 0.f;
    }
  }

  v8f m_a = {}, m_b = {}, e_aa = {}, e_bb = {}, e_ab = {};
#pragma unroll
  for (int c = 0; c < 7; ++c) {
    const int k0   = 4 * c + 2 * lhi;
    const int base = (r0 + k0) * HSTR + j0 + llo;      // r0+k <= 43 < 48, in-bounds
    v2f B0; B0[0] = sH[0][base]; B0[1] = sH[0][base + HSTR];
    v2f B1; B1[0] = sH[1][base]; B1[1] = sH[1][base + HSTR];
    v2f B2; B2[0] = sH[2][base]; B2[1] = sH[2][base + HSTR];
    v2f B3; B3[0] = sH[3][base]; B3[1] = sH[3][base + HSTR];
    v2f B4; B4[0] = sH[4][base]; B4[1] = sH[4][base + HSTR];
    m_a  = __builtin_amdgcn_wmma_f32_16x16x4_f32(false, At[c], false, B0, (short)0, m_a,  false, false);
    m_b  = __builtin_amdgcn_wmma_f32_16x16x4_f32(false, At[c], false, B1, (short)0, m_b,  false, false);
    e_aa = __builtin_amdgcn_wmma_f32_16x16x4_f32(false, At[c], false, B2, (short)0, e_aa, false, false);
    e_bb = __builtin_amdgcn_wmma_f32_16x16x4_f32(false, At[c], false, B3, (short)0, e_bb, false, false);
    e_ab = __builtin_amdgcn_wmma_f32_16x16x4_f32(false, At[c], false, B4, (short)0, e_ab, false, false);
  }

  // ---------- SSIM map + reduction ----------
  const float C1 = 1e-4f;   // 0.01^2
  const float C2 = 9e-4f;   // 0.03^2
  float lsum = 0.f;
#pragma unroll
  for (int v = 0; v < 8; ++v) {
    float ma  = m_a[v], mb = m_b[v];
    float ma2 = ma * ma, mb2 = mb * mb, mab = ma * mb;
    float va  = e_aa[v] - ma2;
    float vb  = e_bb[v] - mb2;
    float vab = e_ab[v] - mab;
    float num = (2.f * mab + C1) * (2.f * vab + C2);
    float den = (ma2 + mb2 + C1) * (va + vb + C2);
    lsum += num / den;
  }
#pragma unroll
  for (int off = 16; off >= 1; off >>= 1)
    lsum += __shfl_xor(lsum, off, 32);
  if (lane == 0) atomicAdd(acc, lsum);
}

__global__ void ssim_fin_kernel(const float* __restrict__ acc, float* __restrict__ out) {
  out[0] = 1.0f - acc[0] * (1.0f / ((float)NBAT * (float)IMG * (float)IMG));
}

extern "C" void kernel_launch(void* const* d_in, const int* in_sizes, int n_in,
                              void* d_out, int out_size, void* d_ws, size_t ws_size,
                              hipStream_t stream) {
  const float* xhat = (const float*)d_in[0];   // x_hat: 32x1x512x512 f32
  const float* x    = (const float*)d_in[1];   // x    : 32x1x512x512 f32
  const float* k2d  = (const float*)d_in[2];   // kernel: 1x1x11x11 f32
  float* out  = (float*)d_out;                 // scalar f32
  float* accp = (float*)d_ws;                  // f32 accumulator in scratch

  ssim_zero_kernel<<<1, 1, 0, stream>>>(accp);
  ssim_main_kernel<<<NBAT * TPI, 128, 0, stream>>>(xhat, x, k2d, accp);
  ssim_fin_kernel<<<1, 1, 0, stream>>>(accp, out);
}